// mean_agg_22617297780832
// MI455X (gfx1250) — compile-verified
//
#include <hip/hip_runtime.h>
#include <math.h>

typedef __attribute__((ext_vector_type(2))) float v2f;
typedef __attribute__((ext_vector_type(8))) float v8f;

#define N_NODES 50000
#define K_NEI   32
#define D_IN    128
#define D_OUT   128
#define M_TILE  32
#define THREADS 256
#define A_STRIDE 516   // 512 + 4 pad -> conflict-free WMMA column reads

__global__ __launch_bounds__(THREADS, 1)
void sage_fused_kernel(const float* __restrict__ fea1,
                       const float* __restrict__ fea2,
                       const int*   __restrict__ idx1,
                       const int*   __restrict__ idx2,
                       const float* __restrict__ W,    // [4*D_IN][D_OUT] row-major
                       float*       __restrict__ out)  // [N][D_OUT]
{
    __shared__ float smemA[M_TILE * A_STRIDE];   // 66 KB of the 320 KB WGP LDS

    const int tid       = threadIdx.x;
    const int blockRow0 = blockIdx.x * M_TILE;

    // ---------------- Phase 1: build concatenated A tile in LDS -------------
    // thread t: row r = t>>3, owns 16 contiguous features (4 x float4)
    {
        const int r    = tid >> 3;          // 0..31
        const int sub  = tid & 7;           // 0..7
        const int c0   = sub * 16;          // feature base within 128
        const int node = blockRow0 + r;
        float* rowA = &smemA[r * A_STRIDE];

        if (node < N_NODES) {
            // direct features: fea1 -> cols [0,128), fea2 -> cols [128,256)
            const float4* f1 = (const float4*)(fea1 + (size_t)node * D_IN + c0);
            const float4* f2 = (const float4*)(fea2 + (size_t)node * D_IN + c0);
            #pragma unroll
            for (int q = 0; q < 4; ++q) {
                *(float4*)&rowA[      c0 + q * 4] = f1[q];
                *(float4*)&rowA[128 + c0 + q * 4] = f2[q];
            }

            // gather-mean over neighbor list 1 -> cols [256,384)
            {
                float4 acc[4];
                #pragma unroll
                for (int q = 0; q < 4; ++q) acc[q] = make_float4(0.f, 0.f, 0.f, 0.f);
                const int* nl = idx1 + (size_t)node * K_NEI;
                for (int j = 0; j < K_NEI; ++j) {
                    const int g = nl[j];
                    const float4* src = (const float4*)(fea1 + (size_t)g * D_IN + c0);
                    #pragma unroll
                    for (int q = 0; q < 4; ++q) {
                        float4 v = src[q];
                        acc[q].x += v.x; acc[q].y += v.y;
                        acc[q].z += v.z; acc[q].w += v.w;
                    }
                }
                const float s = 1.0f / (float)K_NEI;
                #pragma unroll
                for (int q = 0; q < 4; ++q) {
                    float4 v = make_float4(acc[q].x * s, acc[q].y * s,
                                           acc[q].z * s, acc[q].w * s);
                    *(float4*)&rowA[256 + c0 + q * 4] = v;
                }
            }

            // gather-mean over neighbor list 2 -> cols [384,512)
            {
                float4 acc[4];
                #pragma unroll
                for (int q = 0; q < 4; ++q) acc[q] = make_float4(0.f, 0.f, 0.f, 0.f);
                const int* nl = idx2 + (size_t)node * K_NEI;
                for (int j = 0; j < K_NEI; ++j) {
                    const int g = nl[j];
                    const float4* src = (const float4*)(fea1 + (size_t)g * D_IN + c0);
                    #pragma unroll
                    for (int q = 0; q < 4; ++q) {
                        float4 v = src[q];
                        acc[q].x += v.x; acc[q].y += v.y;
                        acc[q].z += v.z; acc[q].w += v.w;
                    }
                }
                const float s = 1.0f / (float)K_NEI;
                #pragma unroll
                for (int q = 0; q < 4; ++q) {
                    float4 v = make_float4(acc[q].x * s, acc[q].y * s,
                                           acc[q].z * s, acc[q].w * s);
                    *(float4*)&rowA[384 + c0 + q * 4] = v;
                }
            }
        } else {
            // pad out-of-range rows with zeros (keeps WMMA inputs defined)
            const float4 z = make_float4(0.f, 0.f, 0.f, 0.f);
            #pragma unroll
            for (int q = 0; q < 4; ++q) {
                *(float4*)&rowA[      c0 + q * 4] = z;
                *(float4*)&rowA[128 + c0 + q * 4] = z;
                *(float4*)&rowA[256 + c0 + q * 4] = z;
                *(float4*)&rowA[384 + c0 + q * 4] = z;
            }
        }
    }

    __syncthreads();

    // ---------------- Phase 2: fp32 WMMA GEMM tile --------------------------
    // wave w: output cols [w*16, w*16+16); M tiles 0 (rows 0-15) and 1 (rows 16-31)
    // A 16x4 f32 fragment: lane l -> m = l&15, k0 = (l>>4)*2, VGPRs hold k0,k0+1
    // B 4x16  f32 fragment: lane l -> n = l&15, VGPRs hold rows k0,k0+1
    const int wave  = tid >> 5;
    const int lane  = tid & 31;
    const int m     = lane & 15;
    const int khalf = (lane >> 4) * 2;             // 0 or 2
    const int ncol  = (lane & 15) + wave * 16;     // global output column

    v8f acc0 = 0.0f;   // rows 0..15 of tile
    v8f acc1 = 0.0f;   // rows 16..31 of tile

    const float* a0p = &smemA[ m        * A_STRIDE + khalf];
    const float* a1p = &smemA[(m + 16)  * A_STRIDE + khalf];
    const float* bp  = W + (size_t)khalf * D_OUT + ncol;

    #pragma unroll 4
    for (int kk = 0; kk < 4 * D_IN; kk += 4) {
        v2f a0; a0.x = a0p[0]; a0.y = a0p[1];
        v2f a1; a1.x = a1p[0]; a1.y = a1p[1];
        v2f b;  b.x  = bp[0];  b.y  = bp[D_OUT];

        acc0 = __builtin_amdgcn_wmma_f32_16x16x4_f32(
                   false, a0, false, b, (short)0, acc0, false, false);
        acc1 = __builtin_amdgcn_wmma_f32_16x16x4_f32(
                   false, a1, false, b, (short)0, acc1, false, false);

        a0p += 4;
        a1p += 4;
        bp  += 4 * D_OUT;
    }

    // ---------------- Epilogue: tanh + store --------------------------------
    // C/D layout: VGPR v -> row (v + (lane>>4)*8) within the 16-row tile
    const int rbase = (lane >> 4) * 8;
    #pragma unroll
    for (int v = 0; v < 8; ++v) {
        const int row0 = blockRow0 + rbase + v;        // M tile 0
        const int row1 = row0 + 16;                    // M tile 1
        if (row0 < N_NODES) out[(size_t)row0 * D_OUT + ncol] = tanhf(acc0[v]);
        if (row1 < N_NODES) out[(size_t)row1 * D_OUT + ncol] = tanhf(acc1[v]);
    }
}

extern "C" void kernel_launch(void* const* d_in, const int* in_sizes, int n_in,
                              void* d_out, int out_size, void* d_ws, size_t ws_size,
                              hipStream_t stream) {
    const float* fea1 = (const float*)d_in[0];
    const float* fea2 = (const float*)d_in[1];
    const int*   idx1 = (const int*)d_in[2];
    const int*   idx2 = (const int*)d_in[3];
    const float* W    = (const float*)d_in[4];
    float*       out  = (float*)d_out;

    const int grid = (N_NODES + M_TILE - 1) / M_TILE;   // 1563 blocks
    sage_fused_kernel<<<grid, THREADS, 0, stream>>>(fea1, fea2, idx1, idx2, W, out);
}